// WaveletTransformLayer_35845797053007
// MI455X (gfx1250) — compile-verified
//
#include <hip/hip_runtime.h>

typedef __attribute__((ext_vector_type(2))) float v2f;
typedef __attribute__((ext_vector_type(8))) float v8f;

#define INV_SQRT2 0.70710678118654752440f

// Haar DWT (db1, level 1) via V_WMMA_F32_16X16X4_F32.
// x: (64, 64, 16384) f32 row-major; out: (64, 2, 64, 8192) f32 row-major.
// One wave handles 4 WMMA tiles = 256 consecutive samples of one (b,c) row.
__global__ __launch_bounds__(256) void haar_wmma_kernel(const float* __restrict__ x,
                                                        float* __restrict__ out) {
  const int lane      = threadIdx.x & 31;
  const int waveInBlk = threadIdx.x >> 5;
  const long long waveId = (long long)blockIdx.x * 8 + waveInBlk;
  const long long base   = waveId * 256;   // first input sample of this wave

  // Decompose flat sample index: base = ((b*64 + c) * 16384) + t0
  const int row = (int)(base >> 14);       // b*64 + c
  const int t0  = (int)(base & 16383);
  const int b   = row >> 6;
  const int ch  = row & 63;
  // out[b][0][c][*] = cA, out[b][1][c][*] = cD
  const long long cAbase = (((long long)b * 2 + 0) * 64 + ch) * 8192 + (t0 >> 1);
  const long long cDbase = cAbase + 64LL * 8192;

  // Constant A-matrix operand (16x4 Haar butterfly), per ISA f32 A layout:
  // lanes 0-15 hold A[M=lane][K=0..1], lanes 16-31 hold A[M=lane-16][K=2..3].
  const int  m  = lane & 15;
  const bool hi = lane >= 16;
  float a0 = 0.0f, a1 = 0.0f;
  if ((!hi && m == 0) || (hi && m == 2)) { a0 = INV_SQRT2; a1 =  INV_SQRT2; }  // cA rows
  if ((!hi && m == 1) || (hi && m == 3)) { a0 = INV_SQRT2; a1 = -INV_SQRT2; }  // cD rows
  v2f a; a.x = a0; a.y = a1;

  // B-matrix (4x16) lane load offset within a 64-sample tile:
  // lane l<16 -> x[4l], x[4l+1]; lane l>=16 -> x[4(l-16)+2], x[4(l-16)+3]
  const int laneOff = (lane & 15) * 4 + (lane >> 4) * 2;

  // Issue all 4 tile loads up front (streaming, non-temporal) for MLP.
  v2f bv[4];
#pragma unroll
  for (int j = 0; j < 4; ++j) {
    bv[j] = __builtin_nontemporal_load(
        (const v2f*)(x + base + (long long)j * 64 + laneOff));
  }

  // 4 WMMAs: D[0][n]=cA[2n], D[1][n]=cD[2n], D[2][n]=cA[2n+1], D[3][n]=cD[2n+1]
  v8f dv[4];
#pragma unroll
  for (int j = 0; j < 4; ++j) {
    v8f c = {};
    dv[j] = __builtin_amdgcn_wmma_f32_16x16x4_f32(
        /*neg_a=*/false, a, /*neg_b=*/false, bv[j],
        /*c_mod=*/(short)0, c, /*reuse_a=*/false, /*reuse_b=*/false);
  }

  // Results live in lanes 0-15 (rows 0-3 of D). Two b64 stores per lane;
  // 16 lanes x 8B = 128 contiguous bytes per store instruction.
  if (lane < 16) {
#pragma unroll
    for (int j = 0; j < 4; ++j) {
      v2f ca; ca.x = dv[j][0]; ca.y = dv[j][2];
      v2f cd; cd.x = dv[j][1]; cd.y = dv[j][3];
      __builtin_nontemporal_store(ca, (v2f*)(out + cAbase + j * 32 + lane * 2));
      __builtin_nontemporal_store(cd, (v2f*)(out + cDbase + j * 32 + lane * 2));
    }
  }
}

extern "C" void kernel_launch(void* const* d_in, const int* in_sizes, int n_in,
                              void* d_out, int out_size, void* d_ws, size_t ws_size,
                              hipStream_t stream) {
  (void)n_in; (void)out_size; (void)d_ws; (void)ws_size;
  const float* x = (const float*)d_in[0];
  float* out     = (float*)d_out;

  const long long n      = (long long)in_sizes[0];   // 64*64*16384 = 67,108,864
  const long long waves  = n / 256;                  // 256 samples per wave
  const long long blocks = waves / 8;                // 8 waves (256 threads) per block

  haar_wmma_kernel<<<dim3((unsigned)blocks), dim3(256), 0, stream>>>(x, out);
}